// HungarianMatcher_32598801776747
// MI455X (gfx1250) — compile-verified
//
#include <hip/hip_runtime.h>
#include <hip/hip_bf16.h>
#include <math.h>

typedef float v2f __attribute__((ext_vector_type(2)));
typedef float v8f __attribute__((ext_vector_type(8)));

#define F_ALPHA 0.25f
#define F_EPS   1e-8f

constexpr int ROWS_TILE = 16;    // rows per block (matches WMMA M)
constexpr int CC_STR    = 84;    // LDS stride for class-cost tile (bank-conflict-free A loads)
constexpr int TGT_MAX   = 1024;
constexpr int TGT_STR   = 9;     // 9 coprime with 64 -> conflict-free target reads
constexpr int BLOCK     = 256;
constexpr int WAVES     = BLOCK / 32;

__global__ __launch_bounds__(BLOCK)
void matcher_cost_kernel(const float* __restrict__ pred_logits,  // [R, NCLS]
                         const float* __restrict__ pred_boxes,   // [R, 4] = t1,t2,c,w
                         const int*   __restrict__ tgt_ids,      // [M]
                         const float* __restrict__ tgt_bbox,     // [M, 4]
                         float* __restrict__ out,                // [R, M]
                         int R, int NCLS, int M)
{
    __shared__ __align__(16) float ccL[ROWS_TILE * CC_STR];   // logits -> focal class cost
    __shared__ __align__(16) float rowD[ROWS_TILE * 8];       // t1,t2,c,w,q1,q2,lenT,pad
    __shared__ __align__(16) float tgtD[TGT_MAX * TGT_STR];   // t1,t2,c,w,q1,q2,lenT,id

    const int tid  = threadIdx.x;
    const int lane = tid & 31;
    const int wave = tid >> 5;
    const int l15  = lane & 15;
    const int half = lane >> 4;
    const int rowBase = blockIdx.x * ROWS_TILE;

    // ---------------- stage logits tile ----------------
    for (int j = tid; j < ROWS_TILE * NCLS; j += BLOCK) {
        int r = j / NCLS, c = j - r * NCLS;
        ccL[r * CC_STR + c] = pred_logits[(size_t)(rowBase + r) * NCLS + c];
    }
    // ---------------- stage row boxes + derived segment data ----------------
    if (tid < ROWS_TILE) {
        const float4 b = *(const float4*)&pred_boxes[(size_t)(rowBase + tid) * 4];
        float* d = &rowD[tid * 8];
        d[0] = b.x; d[1] = b.y; d[2] = b.z; d[3] = b.w;
        d[4] = b.z - 0.5f * b.w;   // cw segment t1
        d[5] = b.z + 0.5f * b.w;   // cw segment t2
        d[6] = b.y - b.x;          // raw segment length
        d[7] = 0.f;
    }
    // ---------------- stage targets + derived ----------------
    for (int n = tid; n < M; n += BLOCK) {
        const float4 b = *(const float4*)&tgt_bbox[(size_t)n * 4];
        float* d = &tgtD[n * TGT_STR];
        d[0] = b.x; d[1] = b.y; d[2] = b.z; d[3] = b.w;
        d[4] = b.z - 0.5f * b.w;
        d[5] = b.z + 0.5f * b.w;
        d[6] = b.y - b.x;
        ((int*)d)[7] = tgt_ids[n];
    }
    __syncthreads();

    // ---------------- in-place focal class cost (softmax over NCLS) ----------------
    // Each wave owns 2 rows; lane covers classes {lane, lane+32, lane+64}.
    for (int rr = 0; rr < ROWS_TILE / WAVES; ++rr) {
        const int r = wave * (ROWS_TILE / WAVES) + rr;
        float* row = &ccL[r * CC_STR];
        const int c1 = lane + 32, c2 = lane + 64;
        float x0 = row[lane];
        float x1 = (c1 < NCLS) ? row[c1] : -1e30f;
        float x2 = (c2 < NCLS) ? row[c2] : -1e30f;
        float m = fmaxf(x0, fmaxf(x1, x2));
        #pragma unroll
        for (int s = 16; s; s >>= 1) m = fmaxf(m, __shfl_xor(m, s, 32));
        float e0 = __expf(x0 - m);
        float e1 = (c1 < NCLS) ? __expf(x1 - m) : 0.f;
        float e2 = (c2 < NCLS) ? __expf(x2 - m) : 0.f;
        float s = e0 + e1 + e2;
        #pragma unroll
        for (int k = 16; k; k >>= 1) s += __shfl_xor(s, k, 32);
        const float rinv = 1.0f / s;
        auto focal = [&](float p) {
            float omp = 1.0f - p;
            float pos = F_ALPHA * omp * omp * (-__logf(p + F_EPS));
            float neg = (1.0f - F_ALPHA) * p * p * (-__logf(omp + F_EPS));
            return pos - neg;
        };
        row[lane] = focal(e0 * rinv);
        if (c1 < NCLS) row[c1] = focal(e1 * rinv);
        if (c2 < NCLS) row[c2] = focal(e2 * rinv);
    }
    __syncthreads();

    // ---------------- main: 16x16 tiles in WMMA C/D layout ----------------
    const int KC = (NCLS + 3) >> 2;          // K chunks of 4 for f32 WMMA
    const int nChunks = M >> 4;
    const float* arow = &ccL[l15 * CC_STR];  // WMMA A: lane l15 supplies row M=l15

    for (int chunk = wave; chunk < nChunks; chunk += WAVES) {
        const int nb = chunk << 4;
        const float* td = &tgtD[(nb + l15) * TGT_STR];  // this lane's target (WMMA col)
        const float tt1 = td[0], tt2 = td[1], tc = td[2], tw = td[3];
        const float tq1 = td[4], tq2 = td[5], tlen = td[6];
        const int   tcls = ((const int*)td)[7];

        v8f acc;
        #pragma unroll
        for (int v = 0; v < 8; ++v) {                     // M = v + 8*half
            const float* rd = &rowD[(v + 8 * half) * 8];  // broadcast read
            float l1 = fabsf(rd[0] - tt1) + fabsf(rd[1] - tt2)
                     + fabsf(rd[2] - tc)  + fabsf(rd[3] - tw);
            // IoU of (c,w)-derived segments
            float lt = fmaxf(rd[4], tq1), rb = fminf(rd[5], tq2);
            float in1 = fmaxf(rb - lt, 0.f);
            float u1  = rd[3] + tw - in1;
            float iou1 = in1 / fmaxf(u1, F_EPS);
            // IoU of raw (t1,t2) segments
            lt = fmaxf(rd[0], tt1); rb = fminf(rd[1], tt2);
            float in2 = fmaxf(rb - lt, 0.f);
            float u2  = rd[6] + tlen - in2;
            float iou2 = in2 / fmaxf(u2, F_EPS);
            acc[v] = l1 - 0.5f * (iou1 + iou2);
        }

        // Class-cost gather on the matrix pipe: acc += ccost(16xNCLS) * onehot(NCLSx16).
        // A layout: lane -> M=l15, VGPR0/1 -> K = 4i+2*half, +1 (exact one-hot f32 matmul).
        auto wstep = [&](int i) {
            const int kk = 4 * i + 2 * half;
            v2f a = *(const v2f*)&arow[kk];
            v2f b;
            b.x = (tcls == kk)     ? 1.0f : 0.0f;
            b.y = (tcls == kk + 1) ? 1.0f : 0.0f;
            acc = __builtin_amdgcn_wmma_f32_16x16x4_f32(false, a, false, b,
                                                        (short)0, acc, false, false);
        };
        if (NCLS == 80) {
            #pragma unroll
            for (int i = 0; i < 20; ++i) wstep(i);
        } else {
            for (int i = 0; i < KC; ++i) wstep(i);
        }

        float* op = &out[(size_t)(rowBase + 8 * half) * M + nb + l15];
        #pragma unroll
        for (int v = 0; v < 8; ++v) op[(size_t)v * M] = acc[v];
    }
}

extern "C" void kernel_launch(void* const* d_in, const int* in_sizes, int n_in,
                              void* d_out, int out_size, void* d_ws, size_t ws_size,
                              hipStream_t stream) {
    const float* pred_logits = (const float*)d_in[0];
    const float* pred_boxes  = (const float*)d_in[1];
    const int*   tgt_ids     = (const int*)d_in[2];
    const float* tgt_bbox    = (const float*)d_in[3];
    float* out = (float*)d_out;

    const int R    = in_sizes[1] / 4;          // bs*Q = 32000
    const int NCLS = in_sizes[0] / R;          // 80
    const int M    = in_sizes[2];              // 1024

    dim3 grid(R / ROWS_TILE);                  // 2000 blocks
    dim3 block(BLOCK);
    matcher_cost_kernel<<<grid, block, 0, stream>>>(pred_logits, pred_boxes,
                                                    tgt_ids, tgt_bbox, out,
                                                    R, NCLS, M);
}